// WatershedFilter_61151744360553
// MI455X (gfx1250) — compile-verified
//
#include <hip/hip_runtime.h>
#include <stdint.h>

#define HH 2048
#define WW 2048
#define NPIX (HH * WW)

typedef float v2f __attribute__((ext_vector_type(2)));
typedef float v8f __attribute__((ext_vector_type(8)));

__device__ __forceinline__ uint32_t f2u(float f) { return __float_as_uint(f); }
__device__ __forceinline__ float u2f(uint32_t u) { return __uint_as_float(u); }

// 1e9f = 0x4E6E6B28 (low 2 bits already 00 -> label 0)
#define INFP_BITS 0x4E6E6B28u

// ---------------------------------------------------------------------------
// scalar slots: sc[0]=gray min bits, sc[1]=gray max bits, sc[2]=label min bits,
// sc[3]=label max bits  (all non-negative floats -> uint compare == float compare)
// ---------------------------------------------------------------------------
__global__ void k_init_scalars(uint32_t* sc) {
  if (threadIdx.x == 0 && blockIdx.x == 0) {
    sc[0] = 0x7F800000u;  // +inf
    sc[1] = 0x00000000u;  // 0.0
    sc[2] = 0x7F800000u;
    sc[3] = 0x00000000u;
  }
}

// ---------------------------------------------------------------------------
// Grayscale einsum via V_WMMA_F32_16X16X4_F32.
// A (16x4): row m = pixel p+m, K = {R,G,B,0}. Per f32 A-layout:
//   lanes 0-15 hold (M=lane, K0) in v0 and (M=lane, K1) in v1
//   lanes 16-31 hold (M=lane-16, K2) in v0 and K3 in v1.
// B (4x16): column n = ITU-R weights down K; same lane split -> built from lane id.
// D[m][n] = gray[p+m] for every n; lanes with col==0 extract 8 rows each.
// Block = 256 threads = 8 waves = 128 pixels.
// ---------------------------------------------------------------------------
__global__ void k_gray_wmma(const float* __restrict__ img,
                            float* __restrict__ rawg,
                            uint32_t* __restrict__ sc) {
  __shared__ float smin[16];
  __shared__ float smax[16];
  const int tid  = threadIdx.x;
  const int lane = tid & 31;
  const int wave = tid >> 5;
  const int col  = lane & 15;
  const bool low = lane < 16;
  const long long p = ((long long)blockIdx.x * 8 + wave) * 16;

  const float* R = img;
  const float* G = img + (size_t)NPIX;
  const float* B = img + (size_t)2 * NPIX;

  v2f a, b;
  if (low) { a.x = R[p + col]; a.y = G[p + col]; b.x = 0.2989f; b.y = 0.587f; }
  else     { a.x = B[p + col]; a.y = 0.0f;       b.x = 0.114f;  b.y = 0.0f;  }
  v8f c = {};
  c = __builtin_amdgcn_wmma_f32_16x16x4_f32(false, a, false, b, (short)0, c,
                                            false, false);

  if (col == 0) {
    const long long base = p + (low ? 0 : 8);
    float mn = c[0], mx = c[0];
#pragma unroll
    for (int i = 0; i < 8; ++i) {
      rawg[base + i] = c[i];
      mn = fminf(mn, c[i]);
      mx = fmaxf(mx, c[i]);
    }
    smin[wave * 2 + (low ? 0 : 1)] = mn;
    smax[wave * 2 + (low ? 0 : 1)] = mx;
  }
  __syncthreads();
  if (tid == 0) {
    float mn = smin[0], mx = smax[0];
#pragma unroll
    for (int i = 1; i < 16; ++i) { mn = fminf(mn, smin[i]); mx = fmaxf(mx, smax[i]); }
    atomicMin(&sc[0], f2u(mn));
    atomicMax(&sc[1], f2u(mx));
  }
}

// ---------------------------------------------------------------------------
// Normalize gray in place, compute markers, write packed seed (dist,label).
// ---------------------------------------------------------------------------
__global__ void k_seed(float* __restrict__ grayio,
                       float* __restrict__ packA,
                       const uint32_t* __restrict__ sc) {
  int i = blockIdx.x * blockDim.x + threadIdx.x;
  if (i >= NPIX) return;
  const float mn = u2f(sc[0]);
  const float mx = u2f(sc[1]);
  const float g = (grayio[i] - mn) / (mx - mn);
  grayio[i] = g;
  uint32_t marker = (g < 0.3f ? 1u : 0u) + (g > 0.7f ? 2u : 0u);
  packA[i] = u2f(marker ? marker : INFP_BITS);  // seeded: dist=0,label; else INF,0
}

// ---------------------------------------------------------------------------
// One directional min-plus sweep (Jacobi from snapshot src -> dst), float4-wide.
// Neighbor of pixel p is p - (DY,DX), matching jnp.roll semantics.
// ---------------------------------------------------------------------------
template <int DY, int DX>
__global__ void k_pass(const float4* __restrict__ src4,
                       float4* __restrict__ dst4,
                       const float4* __restrict__ gray4) {
  int g = blockIdx.x * blockDim.x + threadIdx.x;
  if (g >= NPIX / 4) return;
  const int base = g << 2;
  const int y = base >> 11;
  const int x = base & (WW - 1);

  const float4 me = src4[g];
  const float4 gr = gray4[g];
  const float INFP = u2f(INFP_BITS);

  float vn[4];
  if (DY != 0) {
    const int ny = y - DY;
    if ((unsigned)ny < (unsigned)HH) {
      const float4 t = src4[(ny << 9) | (x >> 2)];
      vn[0] = t.x; vn[1] = t.y; vn[2] = t.z; vn[3] = t.w;
    } else {
      vn[0] = vn[1] = vn[2] = vn[3] = INFP;
    }
  } else if (DX == 1) {           // neighbor at x-1
    vn[1] = me.x; vn[2] = me.y; vn[3] = me.z;
    vn[0] = (x == 0) ? INFP : ((const float*)src4)[base - 1];
  } else {                        // DX == -1, neighbor at x+1
    vn[0] = me.y; vn[1] = me.z; vn[2] = me.w;
    vn[3] = (x + 4 == WW) ? INFP : ((const float*)src4)[base + 4];
  }

  const float mef[4] = {me.x, me.y, me.z, me.w};
  const float grf[4] = {gr.x, gr.y, gr.z, gr.w};
  float r[4];
#pragma unroll
  for (int k = 0; k < 4; ++k) {
    const uint32_t nb = f2u(vn[k]);
    const float cand = u2f(nb & ~3u) + grf[k] + 1.0f;   // nd + gray + compactness
    const uint32_t cb = f2u(cand) & ~3u;
    const float myd = u2f(f2u(mef[k]) & ~3u);
    r[k] = (u2f(cb) < myd) ? u2f(cb | (nb & 3u)) : mef[k];
  }
  dst4[g] = make_float4(r[0], r[1], r[2], r[3]);
}

// ---------------------------------------------------------------------------
// Label min/max reduction over final packed buffer.
// ---------------------------------------------------------------------------
__global__ void k_labminmax(const float* __restrict__ pack, uint32_t* sc) {
  __shared__ float smn[256];
  __shared__ float smx[256];
  const int tid = threadIdx.x;
  const int i = blockIdx.x * blockDim.x + tid;
  float lf = (i < NPIX) ? (float)(f2u(pack[i]) & 3u) : 0.0f;
  smn[tid] = lf;
  smx[tid] = lf;
  __syncthreads();
  for (int s = 128; s > 0; s >>= 1) {
    if (tid < s) {
      smn[tid] = fminf(smn[tid], smn[tid + s]);
      smx[tid] = fmaxf(smx[tid], smx[tid + s]);
    }
    __syncthreads();
  }
  if (tid == 0) {
    atomicMin(&sc[2], f2u(smn[0]));
    atomicMax(&sc[3], f2u(smx[0]));
  }
}

__global__ void k_out(const float* __restrict__ pack,
                      const uint32_t* __restrict__ sc,
                      float* __restrict__ out) {
  int i = blockIdx.x * blockDim.x + threadIdx.x;
  if (i >= NPIX) return;
  const float mn = u2f(sc[2]);
  const float mx = u2f(sc[3]);
  const float lab = (float)(f2u(pack[i]) & 3u);
  out[i] = (lab - mn) / fmaxf(mx - mn, 1e-12f);
}

// ---------------------------------------------------------------------------
extern "C" void kernel_launch(void* const* d_in, const int* in_sizes, int n_in,
                              void* d_out, int out_size, void* d_ws, size_t ws_size,
                              hipStream_t stream) {
  const float* img = (const float*)d_in[0];
  float* ws = (float*)d_ws;
  float* gray  = ws;                          // 16 MB (raw, then normalized)
  float* packA = ws + (size_t)NPIX;           // 16 MB packed dist|label
  float* packB = ws + (size_t)2 * NPIX;       // 16 MB packed dist|label
  uint32_t* sc = (uint32_t*)(ws + (size_t)3 * NPIX);
  float* out = (float*)d_out;

  k_init_scalars<<<1, 64, 0, stream>>>(sc);
  k_gray_wmma<<<NPIX / 128, 256, 0, stream>>>(img, gray, sc);

  const int blk = 256;
  const int grd = (NPIX + blk - 1) / blk;
  k_seed<<<grd, blk, 0, stream>>>(gray, packA, sc);

  const int grd4 = (NPIX / 4 + blk - 1) / blk;
  float* src = packA;
  float* dst = packB;
  for (int it = 0; it < 64; ++it) {
    k_pass<1, 0><<<grd4, blk, 0, stream>>>((const float4*)src, (float4*)dst,
                                           (const float4*)gray);
    { float* t = src; src = dst; dst = t; }
    k_pass<-1, 0><<<grd4, blk, 0, stream>>>((const float4*)src, (float4*)dst,
                                            (const float4*)gray);
    { float* t = src; src = dst; dst = t; }
    k_pass<0, 1><<<grd4, blk, 0, stream>>>((const float4*)src, (float4*)dst,
                                           (const float4*)gray);
    { float* t = src; src = dst; dst = t; }
    k_pass<0, -1><<<grd4, blk, 0, stream>>>((const float4*)src, (float4*)dst,
                                            (const float4*)gray);
    { float* t = src; src = dst; dst = t; }
  }
  // 256 swaps -> final data back in packA (== src)
  k_labminmax<<<grd, blk, 0, stream>>>(src, sc);
  k_out<<<grd, blk, 0, stream>>>(src, sc, out);
}